// SwitchMoe_21500606284005
// MI455X (gfx1250) — compile-verified
//
#include <hip/hip_runtime.h>
#include <hip/hip_bf16.h>

typedef __attribute__((ext_vector_type(2))) float v2f;
typedef __attribute__((ext_vector_type(4))) float v4f;
typedef __attribute__((ext_vector_type(8))) float v8f;

#define B_   8
#define S_   2048
#define D_   1024
#define E_   8
#define CAP_ 512

// ---------------- Router: one wave (32 lanes) per token ----------------
__global__ void __launch_bounds__(256) moe_router(
    const float* __restrict__ x, const float* __restrict__ gw,
    float* __restrict__ out_logits, int* __restrict__ ws_expert,
    float* __restrict__ ws_prob) {
  int gwave = (int)((blockIdx.x * blockDim.x + threadIdx.x) >> 5);
  int lane  = threadIdx.x & 31;
  if (gwave >= B_ * S_) return;
  const float* xr = x + (size_t)gwave * D_;

  float acc[E_];
#pragma unroll
  for (int e = 0; e < E_; ++e) acc[e] = 0.f;

  for (int d = lane * 4; d < D_; d += 32 * 4) {
    v4f xv = *(const v4f*)(xr + d);
#pragma unroll
    for (int e = 0; e < E_; ++e) {
      v4f gv = *(const v4f*)(gw + e * D_ + d);
      acc[e] += xv.x * gv.x + xv.y * gv.y + xv.z * gv.z + xv.w * gv.w;
    }
  }
#pragma unroll
  for (int e = 0; e < E_; ++e) {
#pragma unroll
    for (int off = 16; off >= 1; off >>= 1)
      acc[e] += __shfl_xor(acc[e], off, 32);
  }

  float m = acc[0];
  int arg = 0;
#pragma unroll
  for (int e = 1; e < E_; ++e) {
    if (acc[e] > m) { m = acc[e]; arg = e; }  // first-occurrence argmax
  }
  float sum = 0.f;
#pragma unroll
  for (int e = 0; e < E_; ++e) sum += __expf(acc[e] - m);
  float topp = 1.0f / sum;

  if (lane < E_) out_logits[(size_t)gwave * E_ + lane] = acc[lane];
  if (lane == 0) { ws_expert[gwave] = arg; ws_prob[gwave] = topp; }
}

// -------- Capacity scan: one wave per batch row, ballot prefix-count --------
__global__ void __launch_bounds__(32) moe_capacity(
    const int* __restrict__ ws_expert, int* __restrict__ ws_keep,
    int* __restrict__ ws_counts, int* __restrict__ ws_list,
    float* __restrict__ out_eidx) {
  int b = blockIdx.x;
  int lane = threadIdx.x & 31;
  unsigned lmask = (1u << lane) - 1u;

  int counts[E_];
#pragma unroll
  for (int e = 0; e < E_; ++e) counts[e] = 0;

  for (int c = 0; c < S_; c += 32) {
    int s = c + lane;
    int my_e = ws_expert[b * S_ + s];
    int keep = 0, pos = 0;
#pragma unroll
    for (int xp = 0; xp < E_; ++xp) {
      unsigned m32 = (unsigned)__ballot(my_e == xp);
      if (my_e == xp) {
        pos  = counts[xp] + __popc(m32 & lmask);
        keep = (pos < CAP_);
      }
      counts[xp] += __popc(m32);  // lane-uniform
    }
    ws_keep[b * S_ + s]  = keep;
    out_eidx[b * S_ + s] = keep ? (float)my_e : 0.0f;
    if (keep) ws_list[(b * E_ + my_e) * CAP_ + pos] = s;
  }
  if (lane < E_) {
    int c = counts[lane];
    ws_counts[b * E_ + lane] = (c < CAP_) ? c : CAP_;
  }
}

// ------- Dropped tokens: hidden = top_prob * x (identity path) -------
__global__ void __launch_bounds__(256) moe_passthrough(
    const float* __restrict__ x, const int* __restrict__ ws_keep,
    const float* __restrict__ ws_prob, float* __restrict__ out) {
  int t = blockIdx.x;
  if (ws_keep[t]) return;
  float p = ws_prob[t];
  int d = threadIdx.x * 4;
  v4f xv = *(const v4f*)(x + (size_t)t * D_ + d);
  v4f o;
  o.x = p * xv.x; o.y = p * xv.y; o.z = p * xv.z; o.w = p * xv.w;
  *(v4f*)(out + (size_t)t * D_ + d) = o;
}

// ------- Expert GEMM: gathered rows x W_e via V_WMMA_F32_16X16X4_F32 -------
// 4 waves/block; each wave owns a 16x64 output strip (4 C tiles) so one A
// fragment feeds 4 WMMAs. Block tile = 16 tokens x 256 cols.
__global__ void __launch_bounds__(128) moe_expert_gemm(
    const float* __restrict__ x, const float* __restrict__ w,
    const int* __restrict__ ws_counts, const int* __restrict__ ws_list,
    const float* __restrict__ ws_prob, float* __restrict__ out) {
  int wave = threadIdx.x >> 5;
  int lane = threadIdx.x & 31;
  int be = blockIdx.z;
  int b = be >> 3, e = be & 7;
  int Ne = ws_counts[be];
  int m0 = blockIdx.y * 16;
  if (m0 >= Ne) return;
  int n0 = blockIdx.x * 256 + wave * 64;
  int hi = lane >> 4;   // half-wave selects K pairs {0,1} vs {2,3}
  int lo = lane & 15;

  const int* list = ws_list + be * CAP_;
  int arow = m0 + lo;
  if (arow >= Ne) arow = Ne - 1;  // padded rows computed but never stored
  int tokA = list[arow];

  // A fragment: lane supplies A[M=lo][k+2*hi .. k+2*hi+1] as aligned float2
  const float* xr = x + ((size_t)b * S_ + tokA) * D_ + (size_t)(hi * 2);
  // B fragments: lane supplies W_e[k+2*hi][n] / W_e[k+2*hi+1][n], n = n0+lo+{0,16,32,48}
  const float* wc = w + (size_t)e * D_ * D_ + (size_t)(hi * 2) * D_ + n0 + lo;

  __builtin_prefetch(xr, 0, 0);   // global_prefetch_b8 hints (L2-resident streams)
  __builtin_prefetch(wc, 0, 0);

  v8f c0 = {0.f,0.f,0.f,0.f,0.f,0.f,0.f,0.f};
  v8f c1 = c0, c2 = c0, c3 = c0;

#pragma unroll 2
  for (int k = 0; k < D_; k += 4) {
    v2f a = *(const v2f*)(xr + k);
    const float* wk0 = wc + (size_t)k * D_;
    const float* wk1 = wk0 + D_;
    v2f b0 = { wk0[0],  wk1[0]  };
    v2f b1 = { wk0[16], wk1[16] };
    v2f b2 = { wk0[32], wk1[32] };
    v2f b3 = { wk0[48], wk1[48] };
    c0 = __builtin_amdgcn_wmma_f32_16x16x4_f32(false, a, false, b0, (short)0, c0, false, false);
    c1 = __builtin_amdgcn_wmma_f32_16x16x4_f32(false, a, false, b1, (short)0, c1, false, false);
    c2 = __builtin_amdgcn_wmma_f32_16x16x4_f32(false, a, false, b2, (short)0, c2, false, false);
    c3 = __builtin_amdgcn_wmma_f32_16x16x4_f32(false, a, false, b3, (short)0, c3, false, false);
  }

  // C layout: VGPR r -> M = r (lanes 0-15) / r+8 (lanes 16-31), N = lane&15
#pragma unroll
  for (int r = 0; r < 8; ++r) {
    int mrow = m0 + r + 8 * hi;
    if (mrow < Ne) {
      int tok = list[mrow];
      float p = ws_prob[b * S_ + tok];
      float* op = out + ((size_t)b * S_ + tok) * D_ + n0 + lo;
      op[0]  = p * c0[r];
      op[16] = p * c1[r];
      op[32] = p * c2[r];
      op[48] = p * c3[r];
    }
  }
}

extern "C" void kernel_launch(void* const* d_in, const int* in_sizes, int n_in,
                              void* d_out, int out_size, void* d_ws, size_t ws_size,
                              hipStream_t stream) {
  const float* x  = (const float*)d_in[0];   // [B,S,D]
  const float* gw = (const float*)d_in[1];   // [E,D]
  const float* ew = (const float*)d_in[2];   // [E,D,D]

  float* out_hidden = (float*)d_out;                               // [B,S,D]
  float* out_logits = out_hidden + (size_t)B_ * S_ * D_;           // [B,S,E]
  float* out_eidx   = out_logits + (size_t)B_ * S_ * E_;           // [B,S]

  char* ws = (char*)d_ws;
  int*   ws_expert = (int*)(ws);                      //  64 KB
  int*   ws_keep   = (int*)(ws + (64 << 10));         //  64 KB
  int*   ws_counts = (int*)(ws + (128 << 10));        //  <1 KB (padded)
  int*   ws_list   = (int*)(ws + (129 << 10));        // 128 KB
  float* ws_prob   = (float*)(ws + (257 << 10));      //  64 KB

  moe_router<<<dim3((B_ * S_) / 8), dim3(256), 0, stream>>>(
      x, gw, out_logits, ws_expert, ws_prob);
  moe_capacity<<<dim3(B_), dim3(32), 0, stream>>>(
      ws_expert, ws_keep, ws_counts, ws_list, out_eidx);
  moe_passthrough<<<dim3(B_ * S_), dim3(256), 0, stream>>>(
      x, ws_keep, ws_prob, out_hidden);
  moe_expert_gemm<<<dim3(D_ / 256, CAP_ / 16, B_ * E_), dim3(128), 0, stream>>>(
      x, ew, ws_counts, ws_list, ws_prob, out_hidden);
}